// Extractor_80066780332735
// MI455X (gfx1250) — compile-verified
//
#include <hip/hip_runtime.h>
#include <hip/hip_bf16.h>

typedef float v2f __attribute__((ext_vector_type(2)));
typedef float v8f __attribute__((ext_vector_type(8)));

#define T_TAGS   32
#define SEQ_L    2048
#define BATCH    256
#define CHUNKS   16
#define CHUNK_S  (SEQ_L / CHUNKS)   // 128
#define CRF_START 0
#define CRF_STOP  1
#define NORM_EVERY 4

__device__ __forceinline__ float wave_max32(float x) {
    x = fmaxf(x, __shfl_xor(x, 16, 32));
    x = fmaxf(x, __shfl_xor(x, 8, 32));
    x = fmaxf(x, __shfl_xor(x, 4, 32));
    x = fmaxf(x, __shfl_xor(x, 2, 32));
    x = fmaxf(x, __shfl_xor(x, 1, 32));
    return x;
}
__device__ __forceinline__ float wave_sum32(float x) {
    x += __shfl_xor(x, 16, 32);
    x += __shfl_xor(x, 8, 32);
    x += __shfl_xor(x, 4, 32);
    x += __shfl_xor(x, 2, 32);
    x += __shfl_xor(x, 1, 32);
    return x;
}

// ---------------------------------------------------------------------------
// Phase 1: each wave (32 threads, one block) computes the normalized 32x32
// linear-domain transfer matrix of one (batch, chunk) segment:
//   M_chunk = prod_{l in chunk, mask>0} diag(exp(feat_l - cmax_l)) * exp(trans)
// with an accumulated log-scale so true matrix = exp(ls) * M_chunk.
// The running product lives in LDS row-major; WMMA f32 16x16x4 does the GEMM.
// ---------------------------------------------------------------------------
__global__ __launch_bounds__(32)
void crf_chunk_kernel(const float* __restrict__ efeats,
                      const float* __restrict__ mask,
                      const float* __restrict__ transitions,
                      float* __restrict__ Pout,     // [B*CHUNKS][1024] row-major
                      float* __restrict__ lsout)    // [B*CHUNKS]
{
    __shared__ __align__(16) float ldsP[1024];   // running product, row-major
    __shared__ float ldsS[32];                   // per-step row scales

    const int lane = threadIdx.x;       // 0..31
    const int task = blockIdx.x;        // b*CHUNKS + c
    const int b    = task / CHUNKS;
    const int c    = task % CHUNKS;
    const int h    = lane >> 4;         // lane half (0/1)
    const int lm   = lane & 15;

    // E = exp(transitions), preloaded as constant A-layout fragments.
    // A (16x4 f32): lane = M (mod 16), VGPR v + 2*half = K.
    // eA[Mi][kc] element v holds E[16*Mi + lm][4*kc + v + 2*h].
    v2f eA[2][8];
#pragma unroll
    for (int Mi = 0; Mi < 2; ++Mi) {
#pragma unroll
        for (int kc = 0; kc < 8; ++kc) {
            const int row = 16 * Mi + lm;
            const int k0  = 4 * kc + 2 * h;
            eA[Mi][kc].x = __expf(transitions[row * 32 + k0 + 0]);
            eA[Mi][kc].y = __expf(transitions[row * 32 + k0 + 1]);
        }
    }

    // Init running product = Identity.
#pragma unroll
    for (int j = 0; j < 32; ++j)
        ldsP[j * 32 + lane] = (j == lane) ? 1.0f : 0.0f;
    __syncthreads();   // single-wave WG: lowers to S_NOP, keeps compiler order

    float logscale = 0.0f;
    int   nn = 0;
    const float* featbase = efeats + (size_t)b * SEQ_L * 32;
    const float* maskbase = mask + (size_t)b * SEQ_L;
    const int l0 = c * CHUNK_S;

    for (int s = 0; s < CHUNK_S; ++s) {
        const int l = l0 + s;
        const float mv = maskbase[l];
        if (mv > 0.0f) {
            // --- per-step row scales s_i = exp(feat_i - cmax) ---
            const float f = featbase[(size_t)l * 32 + lane];
            const float cmax = wave_max32(f);
            ldsS[lane] = __expf(f - cmax);
            __syncthreads();

            float scl0[8], scl1[8];   // row scales for D rows (C layout: M = r + 8h)
#pragma unroll
            for (int r = 0; r < 8; ++r) {
                scl0[r] = ldsS[r + 8 * h];        // Mi = 0
                scl1[r] = ldsS[16 + r + 8 * h];   // Mi = 1
            }

            // --- D = E * P  (32x32x32 via 16x16x4 WMMA tiles) ---
            v8f acc00 = {}, acc01 = {}, acc10 = {}, acc11 = {};
#pragma unroll
            for (int kc = 0; kc < 8; ++kc) {
                // B (4x16 f32): lane = N (mod 16), VGPR v + 2*half = K.
                // B tile (kc,Ni) element v = P[4kc + v + 2h][16Ni + lm]
                const int r0 = (4 * kc + 2 * h) * 32 + lm;
                v2f b0, b1;
                b0.x = ldsP[r0 + 0];        b0.y = ldsP[r0 + 32];        // Ni = 0
                b1.x = ldsP[r0 + 16];       b1.y = ldsP[r0 + 32 + 16];   // Ni = 1
                acc00 = __builtin_amdgcn_wmma_f32_16x16x4_f32(
                            false, eA[0][kc], false, b0, (short)0, acc00, false, false);
                acc01 = __builtin_amdgcn_wmma_f32_16x16x4_f32(
                            false, eA[0][kc], false, b1, (short)0, acc01, false, false);
                acc10 = __builtin_amdgcn_wmma_f32_16x16x4_f32(
                            false, eA[1][kc], false, b0, (short)0, acc10, false, false);
                acc11 = __builtin_amdgcn_wmma_f32_16x16x4_f32(
                            false, eA[1][kc], false, b1, (short)0, acc11, false, false);
            }

            // --- apply diag(exp(feat - cmax)) to rows of D ---
#pragma unroll
            for (int r = 0; r < 8; ++r) {
                acc00[r] *= scl0[r];  acc01[r] *= scl0[r];
                acc10[r] *= scl1[r];  acc11[r] *= scl1[r];
            }
            logscale += cmax;

            // --- periodic renormalization (keeps f32 in range) ---
            if (++nn >= NORM_EVERY) {
                nn = 0;
                float pm = 0.0f;
#pragma unroll
                for (int r = 0; r < 8; ++r) {
                    pm = fmaxf(pm, acc00[r]); pm = fmaxf(pm, acc01[r]);
                    pm = fmaxf(pm, acc10[r]); pm = fmaxf(pm, acc11[r]);
                }
                pm = wave_max32(pm);
                logscale += __logf(pm);
                const float inv = 1.0f / pm;
#pragma unroll
                for (int r = 0; r < 8; ++r) {
                    acc00[r] *= inv; acc01[r] *= inv;
                    acc10[r] *= inv; acc11[r] *= inv;
                }
            }

            // --- store D back to LDS row-major (C layout: M = 16Mi + r + 8h) ---
            __syncthreads();
#pragma unroll
            for (int r = 0; r < 8; ++r) {
                const int row0 = (r + 8 * h) * 32;          // Mi = 0
                const int row1 = (16 + r + 8 * h) * 32;     // Mi = 1
                ldsP[row0 + lm]      = acc00[r];
                ldsP[row0 + 16 + lm] = acc01[r];
                ldsP[row1 + lm]      = acc10[r];
                ldsP[row1 + 16 + lm] = acc11[r];
            }
            __syncthreads();
        }
    }

    // Write chunk matrix + logscale to workspace.
    float4* dst = (float4*)(Pout + (size_t)task * 1024);
    const float4* src = (const float4*)ldsP;
#pragma unroll
    for (int j = 0; j < 8; ++j)
        dst[lane + j * 32] = src[lane + j * 32];
    if (lane == 0) lsout[task] = logscale;
}

// ---------------------------------------------------------------------------
// Phase 2: one wave per batch folds chunk matrices into the one-hot START
// vector (linear domain, renormalized), then applies the STOP transition row.
// ---------------------------------------------------------------------------
__global__ __launch_bounds__(32)
void crf_combine_kernel(const float* __restrict__ Pmats,
                        const float* __restrict__ lschunks,
                        const float* __restrict__ transitions,
                        float* __restrict__ fwd_out)
{
    const int b = blockIdx.x;
    const int lane = threadIdx.x;

    float v  = (lane == CRF_START) ? 1.0f : 0.0f;   // exp(-10000) == 0, exp(0) == 1
    float LS = 0.0f;

    for (int c = 0; c < CHUNKS; ++c) {
        const float* P = Pmats + ((size_t)(b * CHUNKS + c)) * 1024;
        const float4* Prow = (const float4*)(P + lane * 32);
        float acc = 0.0f;
#pragma unroll
        for (int q = 0; q < 8; ++q) {
            const float4 t4 = Prow[q];
            acc += t4.x * __shfl(v, 4 * q + 0, 32);
            acc += t4.y * __shfl(v, 4 * q + 1, 32);
            acc += t4.z * __shfl(v, 4 * q + 2, 32);
            acc += t4.w * __shfl(v, 4 * q + 3, 32);
        }
        const float m = wave_max32(acc);
        v = acc / m;
        LS += __logf(m) + lschunks[b * CHUNKS + c];
    }

    const float term = v * __expf(transitions[CRF_STOP * 32 + lane]);
    const float tot = wave_sum32(term);
    if (lane == 0) fwd_out[b] = __logf(tot) + LS;
}

// ---------------------------------------------------------------------------
// Phase 3: gold score per batch (masked gather-sum + STOP transition).
// ---------------------------------------------------------------------------
__global__ __launch_bounds__(256)
void crf_gold_kernel(const float* __restrict__ efeats,
                     const float* __restrict__ mask,
                     const int* __restrict__ tgt,
                     const float* __restrict__ transitions,
                     float* __restrict__ gold_out)
{
    __shared__ float red[256];
    __shared__ float redm[256];
    const int b = blockIdx.x;
    const int t = threadIdx.x;

    float sum = 0.0f, msum = 0.0f;
    for (int l = t; l < SEQ_L; l += 256) {
        const float mv = mask[(size_t)b * SEQ_L + l];
        const int cur  = tgt[(size_t)b * SEQ_L + l];
        const int prev = (l == 0) ? CRF_START : tgt[(size_t)b * SEQ_L + l - 1];
        const float sc = transitions[cur * 32 + prev]
                       + efeats[((size_t)b * SEQ_L + l) * 32 + cur];
        sum  += (mv > 0.0f) ? sc : 0.0f;
        msum += mv;
    }
    red[t] = sum; redm[t] = msum;
    __syncthreads();
    for (int off = 128; off > 0; off >>= 1) {
        if (t < off) { red[t] += red[t + off]; redm[t] += redm[t + off]; }
        __syncthreads();
    }
    if (t == 0) {
        const int len = (int)(redm[0] + 0.5f);
        const int last = (len > 0) ? tgt[(size_t)b * SEQ_L + len - 1] : CRF_START;
        gold_out[b] = red[0] + transitions[CRF_STOP * 32 + last];
    }
}

// ---------------------------------------------------------------------------
// Phase 4: loss = sum_b(forward - gold) / B
// ---------------------------------------------------------------------------
__global__ __launch_bounds__(256)
void crf_final_kernel(const float* __restrict__ fwd,
                      const float* __restrict__ gold,
                      float* __restrict__ out)
{
    __shared__ float red[256];
    const int t = threadIdx.x;
    red[t] = fwd[t] - gold[t];   // BATCH == 256
    __syncthreads();
    for (int off = 128; off > 0; off >>= 1) {
        if (t < off) red[t] += red[t + off];
        __syncthreads();
    }
    if (t == 0) out[0] = red[0] / (float)BATCH;
}

extern "C" void kernel_launch(void* const* d_in, const int* in_sizes, int n_in,
                              void* d_out, int out_size, void* d_ws, size_t ws_size,
                              hipStream_t stream) {
    const float* efeats      = (const float*)d_in[0];   // (B, L, T) f32
    const float* mask        = (const float*)d_in[1];   // (B, L)    f32
    const int*   tgt         = (const int*)d_in[2];     // (B, L)    i32
    const float* transitions = (const float*)d_in[3];   // (T, T)    f32
    float* out = (float*)d_out;

    // Workspace layout (floats): ~16.8 MB total.
    float* wsP  = (float*)d_ws;                         // B*CHUNKS*1024
    float* wsLS = wsP + (size_t)BATCH * CHUNKS * 1024;  // B*CHUNKS
    float* wsF  = wsLS + (size_t)BATCH * CHUNKS;        // B
    float* wsG  = wsF + BATCH;                          // B

    crf_chunk_kernel<<<BATCH * CHUNKS, 32, 0, stream>>>(efeats, mask, transitions,
                                                        wsP, wsLS);
    crf_combine_kernel<<<BATCH, 32, 0, stream>>>(wsP, wsLS, transitions, wsF);
    crf_gold_kernel<<<BATCH, 256, 0, stream>>>(efeats, mask, tgt, transitions, wsG);
    crf_final_kernel<<<1, 256, 0, stream>>>(wsF, wsG, out);
}